// SumLayer_6047313953253
// MI455X (gfx1250) — compile-verified
//
#include <hip/hip_runtime.h>
#include <hip/hip_bf16.h>
#include <math.h>

// -----------------------------------------------------------------------------
// Segmented log-sum-exp over gathered values (sum-layer forward), MI455X.
//   xg  = x[ptrs]                         [E]
//   m   = segment_max(xg, csr)            [NSEG]  (empty segments -> 0)
//   s   = segment_sum(exp(xg - m[csr]))   [NSEG]  (+ EPSILON)
//   out = log(s) + m
//
// csr is SORTED. Each thread owns 4 consecutive edges (b128 non-temporal
// stream loads). Per-thread reduce-by-key flushes interior runs directly;
// head/tail partials are stitched across lanes with a wave32 segmented
// shuffle scan, so roughly one atomic per run (avg run length 4) reaches L2.
//
// Memory reasoning (23.3 TB/s HBM, 192 MB L2): x (16 MB), m_enc+s (64 MB),
// out (32 MB) are L2-resident; ptrs/csr (512 MB over two passes) stream
// through with non-temporal hints so they never evict the working set.
// -----------------------------------------------------------------------------

typedef int      v4i __attribute__((ext_vector_type(4)));
typedef unsigned v4u __attribute__((ext_vector_type(4)));
typedef float    v4f __attribute__((ext_vector_type(4)));

#define EPSILON_F   1e-15f
#define ENC_NEG_INF 0x007FFFFFu   // enc(-inf) = ~bits(0xFF800000)
#define SEG_SENT    0x7FFFFFFF    // sentinel segment id for invalid edges

// Monotonic order-preserving mapping float -> uint32 (for atomic max).
__device__ __forceinline__ unsigned enc_f32(float f) {
    unsigned u = __float_as_uint(f);
    return (u & 0x80000000u) ? ~u : (u | 0x80000000u);
}
__device__ __forceinline__ float dec_f32(unsigned e) {
    unsigned u = (e & 0x80000000u) ? (e ^ 0x80000000u) : ~e;
    return __uint_as_float(u);
}

__global__ void k_init(unsigned* __restrict__ m_enc,
                       float* __restrict__ s, int nseg) {
    int t = blockIdx.x * blockDim.x + threadIdx.x;
    int base = t * 4;
    if (base + 3 < nseg) {
        ((v4u*)m_enc)[t] = (v4u)(ENC_NEG_INF);
        ((v4f*)s)[t]     = (v4f)(0.0f);
    } else {
        for (int k = 0; k < 4; ++k) {
            int i = base + k;
            if (i < nseg) { m_enc[i] = ENC_NEG_INF; s[i] = 0.0f; }
        }
    }
}

// ---- Pass 1: per-segment max -----------------------------------------------
__global__ void k_seg_max(const float* __restrict__ x,
                          const int* __restrict__ ptrs,
                          const int* __restrict__ csr,
                          unsigned* __restrict__ m_enc, int E) {
    const int t    = blockIdx.x * blockDim.x + threadIdx.x;
    const int lane = threadIdx.x & 31;
    const int base = t * 4;

    int   sg[4];
    float v[4];
    if (base + 3 < E) {
        v4i p = __builtin_nontemporal_load(&((const v4i*)ptrs)[t]);
        v4i c = __builtin_nontemporal_load(&((const v4i*)csr)[t]);
        sg[0] = c.x; sg[1] = c.y; sg[2] = c.z; sg[3] = c.w;
        v[0] = x[p.x]; v[1] = x[p.y]; v[2] = x[p.z]; v[3] = x[p.w];
    } else {
        #pragma unroll
        for (int k = 0; k < 4; ++k) {
            int i = base + k;
            if (i < E) { sg[k] = csr[i]; v[k] = x[ptrs[i]]; }
            else       { sg[k] = SEG_SENT; v[k] = -INFINITY; }
        }
    }

    // Serial reduce-by-key over the 4 owned edges.
    int   curSeg  = sg[0];
    float curAgg  = v[0];
    int   headSeg = sg[0];
    float headAgg = -INFINITY;
    bool  multi   = false;
    #pragma unroll
    for (int k = 1; k < 4; ++k) {
        if (sg[k] == curSeg) {
            curAgg = fmaxf(curAgg, v[k]);
        } else {
            if (!multi) { headAgg = curAgg; multi = true; }       // defer head
            else if (curSeg != SEG_SENT)                          // interior
                atomicMax(&m_enc[curSeg], enc_f32(curAgg));
            curSeg = sg[k]; curAgg = v[k];
        }
    }
    const int tailSeg = curSeg;

    // Wave32 segmented inclusive max-scan over tail partials.
    float a = curAgg;
    #pragma unroll
    for (int d = 1; d < 32; d <<= 1) {
        float oa = __shfl_up(a, d, 32);
        int   os = __shfl_up(tailSeg, d, 32);
        if (lane >= d && os == tailSeg) a = fmaxf(a, oa);
    }
    float prevScan = __shfl_up(a, 1, 32);
    int   prevTail = __shfl_up(tailSeg, 1, 32);
    int   nextHead = __shfl_down(headSeg, 1, 32);

    // Head-run flush (multi-run lanes own the run ending inside them).
    bool conn = (lane > 0) && (prevTail == headSeg);
    if (multi && headSeg != SEG_SENT) {
        float h = conn ? fmaxf(headAgg, prevScan) : headAgg;
        atomicMax(&m_enc[headSeg], enc_f32(h));
    }
    // Tail-run flush iff the run does not continue into the next lane.
    bool tailEnd = (lane == 31) || (nextHead != tailSeg);
    if (tailEnd && tailSeg != SEG_SENT) {
        atomicMax(&m_enc[tailSeg], enc_f32(a));
    }
}

// ---- Pass 2: segmented sum of exp(v - m[seg]) ------------------------------
__global__ void k_seg_sum(const float* __restrict__ x,
                          const int* __restrict__ ptrs,
                          const int* __restrict__ csr,
                          const unsigned* __restrict__ m_enc,
                          float* __restrict__ s, int E) {
    const int t    = blockIdx.x * blockDim.x + threadIdx.x;
    const int lane = threadIdx.x & 31;
    const int base = t * 4;

    int   sg[4];
    float e[4];
    if (base + 3 < E) {
        v4i p = __builtin_nontemporal_load(&((const v4i*)ptrs)[t]);
        v4i c = __builtin_nontemporal_load(&((const v4i*)csr)[t]);
        sg[0] = c.x; sg[1] = c.y; sg[2] = c.z; sg[3] = c.w;
        e[0] = x[p.x]; e[1] = x[p.y]; e[2] = x[p.z]; e[3] = x[p.w];
        #pragma unroll
        for (int k = 0; k < 4; ++k) {
            float m = dec_f32(m_enc[sg[k]]);   // seg touched in pass 1
            float r = expf(e[k] - m);
            e[k] = (r != r) ? 0.0f : r;        // nan_to_num(nan=0.0)
        }
    } else {
        #pragma unroll
        for (int k = 0; k < 4; ++k) {
            int i = base + k;
            if (i < E) {
                sg[k] = csr[i];
                float m = dec_f32(m_enc[sg[k]]);
                float r = expf(x[ptrs[i]] - m);
                e[k] = (r != r) ? 0.0f : r;
            } else { sg[k] = SEG_SENT; e[k] = 0.0f; }
        }
    }

    // Serial reduce-by-key over the 4 owned edges.
    int   curSeg  = sg[0];
    float curAgg  = e[0];
    int   headSeg = sg[0];
    float headAgg = 0.0f;
    bool  multi   = false;
    #pragma unroll
    for (int k = 1; k < 4; ++k) {
        if (sg[k] == curSeg) {
            curAgg += e[k];
        } else {
            if (!multi) { headAgg = curAgg; multi = true; }
            else if (curSeg != SEG_SENT)
                atomicAdd(&s[curSeg], curAgg);
            curSeg = sg[k]; curAgg = e[k];
        }
    }
    const int tailSeg = curSeg;

    // Wave32 segmented inclusive sum-scan over tail partials.
    float a = curAgg;
    #pragma unroll
    for (int d = 1; d < 32; d <<= 1) {
        float oa = __shfl_up(a, d, 32);
        int   os = __shfl_up(tailSeg, d, 32);
        if (lane >= d && os == tailSeg) a += oa;
    }
    float prevScan = __shfl_up(a, 1, 32);
    int   prevTail = __shfl_up(tailSeg, 1, 32);
    int   nextHead = __shfl_down(headSeg, 1, 32);

    bool conn = (lane > 0) && (prevTail == headSeg);
    if (multi && headSeg != SEG_SENT) {
        float h = conn ? (headAgg + prevScan) : headAgg;
        atomicAdd(&s[headSeg], h);
    }
    bool tailEnd = (lane == 31) || (nextHead != tailSeg);
    if (tailEnd && tailSeg != SEG_SENT) {
        atomicAdd(&s[tailSeg], a);
    }
}

// ---- Pass 3: out = log(s + eps) + m ----------------------------------------
__global__ void k_finalize(const unsigned* __restrict__ m_enc,
                           const float* __restrict__ s,
                           float* __restrict__ out, int nseg) {
    int t = blockIdx.x * blockDim.x + threadIdx.x;
    int base = t * 4;
    if (base + 3 < nseg) {
        v4u me = ((const v4u*)m_enc)[t];
        v4f sv = ((const v4f*)s)[t];
        v4f r;
        #pragma unroll
        for (int k = 0; k < 4; ++k) {
            unsigned e = me[k];
            float m = (e == ENC_NEG_INF) ? 0.0f : dec_f32(e);
            r[k] = logf(sv[k] + EPSILON_F) + m;
        }
        __builtin_nontemporal_store(r, &((v4f*)out)[t]);
    } else {
        for (int k = 0; k < 4; ++k) {
            int i = base + k;
            if (i < nseg) {
                unsigned e = m_enc[i];
                float m = (e == ENC_NEG_INF) ? 0.0f : dec_f32(e);
                out[i] = logf(s[i] + EPSILON_F) + m;
            }
        }
    }
}

extern "C" void kernel_launch(void* const* d_in, const int* in_sizes, int n_in,
                              void* d_out, int out_size, void* d_ws, size_t ws_size,
                              hipStream_t stream) {
    const float* x    = (const float*)d_in[0];   // [N_IN]
    const int*   ptrs = (const int*)d_in[1];     // [E]
    const int*   csr  = (const int*)d_in[2];     // [E]
    float*       out  = (float*)d_out;           // [NSEG]

    const int E    = in_sizes[1];
    const int NSEG = out_size;

    // Workspace: [NSEG] u32 encoded max | [NSEG] f32 sum  (64 MB total)
    unsigned* m_enc = (unsigned*)d_ws;
    float*    s     = (float*)((char*)d_ws + (size_t)NSEG * sizeof(unsigned));

    const int BLK = 256;
    const int EPT = 4;                              // edges / thread
    const int gSeg  = (NSEG + BLK * EPT - 1) / (BLK * EPT);
    const int gEdge = (E    + BLK * EPT - 1) / (BLK * EPT);

    k_init    <<<gSeg,  BLK, 0, stream>>>(m_enc, s, NSEG);
    k_seg_max <<<gEdge, BLK, 0, stream>>>(x, ptrs, csr, m_enc, E);
    k_seg_sum <<<gEdge, BLK, 0, stream>>>(x, ptrs, csr, m_enc, s, E);
    k_finalize<<<gSeg,  BLK, 0, stream>>>(m_enc, s, out, NSEG);
}